// TensoRFEncoder_28630251995601
// MI455X (gfx1250) — compile-verified
//
#include <hip/hip_runtime.h>
#include <math.h>

// TensoRF appearance encoder for MI455X (gfx1250).
// Memory-bound gather/interp workload: ~240 MB HBM traffic/call -> ~10 us floor
// at 23.3 TB/s. Tables (786 KB planes + 6 KB lines) are L2/WGP$-resident.
// WMMA does not apply (no shared reduction dimension); optimize data movement:
//  - repack plane texels to float4 -> b128 corner gathers (4x fewer vmem ops)
//  - line table staged to LDS via gfx1250 async-to-LDS path when available
//  - nontemporal b128 stores for the 192 MB output stream
//  - global_prefetch_b8 on the coordinate stream

typedef __attribute__((ext_vector_type(4))) float v4f;
typedef __attribute__((address_space(1))) int g_int;   // global (device) dword
typedef __attribute__((address_space(3))) int l_int;   // LDS dword

#define TG     128   // grid resolution G
#define APPC   4     // channels used
#define NCOMP  5     // channels stored
#define NPLANE 3

#if defined(__has_builtin)
#if __has_builtin(__builtin_amdgcn_global_load_async_to_lds_b32)
#define HAVE_ASYNC_LDS 1
#endif
#endif

// ---------------------------------------------------------------------------
// Repack plane_coef [p][c][y][x] (c-stride 64KB) -> [p][y][x][c] float4 texels
// so each bilinear corner is a single b128 load.
// ---------------------------------------------------------------------------
__global__ __launch_bounds__(256) void repack_planes_kernel(
    const float* __restrict__ plane, v4f* __restrict__ packed)
{
    const int total = NPLANE * TG * TG;
    int i = blockIdx.x * blockDim.x + threadIdx.x;
    if (i >= total) return;
    int p  = i / (TG * TG);
    int yx = i - p * (TG * TG);
    const float* b = plane + (size_t)(p * NCOMP) * (TG * TG) + yx;
    v4f v;
    v.x = b[0 * TG * TG];
    v.y = b[1 * TG * TG];
    v.z = b[2 * TG * TG];
    v.w = b[3 * TG * TG];
    packed[i] = v;
}

// ---------------------------------------------------------------------------
// Main kernel. PACKED=true reads float4 texels from d_ws; PACKED=false reads
// the original strided layout (fallback if workspace is too small).
// ---------------------------------------------------------------------------
template <bool PACKED>
__global__ __launch_bounds__(256) void tensorf_kernel(
    const float* __restrict__ x,
    const float* __restrict__ plane,    // original layout (fallback path)
    const v4f*  __restrict__ planes4,   // packed texels (fast path)
    const float* __restrict__ line,
    float* __restrict__ out,
    int n)
{
    // Stage line coefficients in LDS as float4 per (plane, y): 6 KB.
    // lline[p*TG + y] holds channels 0..3 for (plane p, grid row y).
    __shared__ v4f lline[NPLANE * TG];

#if defined(HAVE_ASYNC_LDS)
    {
        // gfx1250 async memory->LDS copy (ASYNCcnt-tracked); each lane moves
        // one dword straight into LDS without touching VGPRs.
        float* lds_f = (float*)lline;
        for (int i = threadIdx.x; i < NPLANE * APPC * TG; i += blockDim.x) {
            int p  = i / (APPC * TG);
            int r  = i - p * (APPC * TG);
            int ch = r / TG;
            int y  = r - ch * TG;
            int ldsw = (p * TG + y) * 4 + ch;          // dword index in lline
            int gw   = (p * NCOMP + ch) * TG + y;      // dword index in line
            __builtin_amdgcn_global_load_async_to_lds_b32(
                (g_int*)(line + gw),
                (l_int*)(lds_f + ldsw),
                0, 0);
        }
#if __has_builtin(__builtin_amdgcn_s_wait_asynccnt)
        __builtin_amdgcn_s_wait_asynccnt(0);
#else
        asm volatile("s_wait_asynccnt 0" ::: "memory");
#endif
    }
#else
    for (int i = threadIdx.x; i < NPLANE * TG; i += blockDim.x) {
        int p = i / TG;
        int y = i - p * TG;
        const float* b = line + (size_t)(p * NCOMP) * TG + y;
        v4f v;
        v.x = b[0 * TG];
        v.y = b[1 * TG];
        v.z = b[2 * TG];
        v.w = b[3 * TG];
        lline[i] = v;
    }
#endif
    __syncthreads();

    const int stride = gridDim.x * blockDim.x;
    for (int idx = blockIdx.x * blockDim.x + threadIdx.x; idx < n; idx += stride) {
        // gfx1250 global_prefetch_b8 on next iteration's coordinate read
        // (speculative: OOB prefetch is dropped silently).
        __builtin_prefetch(x + (size_t)(idx + stride) * 3, 0, 1);

        float c[3];
        c[0] = x[(size_t)idx * 3 + 0];
        c[1] = x[(size_t)idx * 3 + 1];
        c[2] = x[(size_t)idx * 3 + 2];

        // Per-coordinate interpolation setup (shared across planes/lines).
        float w[3];
        int a0[3], a1[3];
#pragma unroll
        for (int i = 0; i < 3; ++i) {
            float f  = (c[i] + 1.0f) * 0.5f * (float)(TG - 1);
            float fl = floorf(f);
            int q0 = (int)fl;
            q0 = q0 < 0 ? 0 : (q0 > TG - 1 ? TG - 1 : q0);
            int q1 = q0 + 1;
            q1 = q1 > TG - 1 ? TG - 1 : q1;
            w[i]  = f - fl;
            a0[i] = q0;
            a1[i] = q1;
        }

        float res[3 * APPC];
#pragma unroll
        for (int p = 0; p < NPLANE; ++p) {
            // plane p: gx = coord[(0,0,1)[p]] (W dim), gy = coord[(1,2,2)[p]] (H dim)
            const int cx = (p == 2) ? 1 : 0;
            const int cy = (p == 0) ? 1 : 2;
            const int cl = 2 - p;   // line coord: (2,1,0)[p]

            float wx = w[cx], wy = w[cy];
            int x0 = a0[cx], x1 = a1[cx];
            int y0 = a0[cy], y1 = a1[cy];

            float w00 = (1.0f - wx) * (1.0f - wy);
            float w10 = wx * (1.0f - wy);
            float w01 = (1.0f - wx) * wy;
            float w11 = wx * wy;

            v4f v00, v10, v01, v11;
            if (PACKED) {
                const v4f* base = planes4 + (size_t)p * TG * TG;
                v00 = base[y0 * TG + x0];
                v10 = base[y0 * TG + x1];
                v01 = base[y1 * TG + x0];
                v11 = base[y1 * TG + x1];
            } else {
#pragma unroll
                for (int ch = 0; ch < APPC; ++ch) {
                    const float* base =
                        plane + (size_t)(p * NCOMP + ch) * (TG * TG);
                    v00[ch] = base[y0 * TG + x0];
                    v10[ch] = base[y0 * TG + x1];
                    v01[ch] = base[y1 * TG + x0];
                    v11[ch] = base[y1 * TG + x1];
                }
            }
            v4f pf = v00 * w00 + v10 * w10 + v01 * w01 + v11 * w11;

            // Line interpolation from LDS (two ds_load_b128).
            float wl = w[cl];
            v4f l0 = lline[p * TG + a0[cl]];
            v4f l1 = lline[p * TG + a1[cl]];
            v4f lf = l0 * (1.0f - wl) + l1 * wl;

            v4f r = pf * lf;
            res[p * APPC + 0] = r.x;
            res[p * APPC + 1] = r.y;
            res[p * APPC + 2] = r.z;
            res[p * APPC + 3] = r.w;
        }

        // 48 B/point streaming output: non-temporal b128 stores.
        v4f o0 = {res[0], res[1], res[2],  res[3]};
        v4f o1 = {res[4], res[5], res[6],  res[7]};
        v4f o2 = {res[8], res[9], res[10], res[11]};
        v4f* op = (v4f*)(out + (size_t)idx * 12);
        __builtin_nontemporal_store(o0, op + 0);
        __builtin_nontemporal_store(o1, op + 1);
        __builtin_nontemporal_store(o2, op + 2);
    }
}

extern "C" void kernel_launch(void* const* d_in, const int* in_sizes, int n_in,
                              void* d_out, int out_size, void* d_ws, size_t ws_size,
                              hipStream_t stream)
{
    (void)n_in; (void)out_size;
    const float* x     = (const float*)d_in[0];
    const float* plane = (const float*)d_in[1];
    const float* line  = (const float*)d_in[2];
    float* out = (float*)d_out;
    const int n = in_sizes[0] / 3;

    const size_t packed_bytes = (size_t)NPLANE * TG * TG * sizeof(v4f); // 768 KB
    const int main_blocks = 2048;   // grid-stride: ~8 points/thread at N=4M

    if (ws_size >= packed_bytes) {
        v4f* planes4 = (v4f*)d_ws;
        const int total = NPLANE * TG * TG;
        repack_planes_kernel<<<(total + 255) / 256, 256, 0, stream>>>(plane, planes4);
        tensorf_kernel<true><<<main_blocks, 256, 0, stream>>>(
            x, plane, planes4, line, out, n);
    } else {
        tensorf_kernel<false><<<main_blocks, 256, 0, stream>>>(
            x, plane, (const v4f*)nullptr, line, out, n);
    }
}